// LSTMwithDones_14156212208141
// MI455X (gfx1250) — compile-verified
//
#include <hip/hip_runtime.h>
#include <math.h>

// LSTM-with-dones for MI455X (gfx1250, wave32).
//
// Per-step fused kernel: gates = [x_t | m*h_{t-1}] @ [W_ih|W_hh]^T + (b_ih+b_hh)
// via V_WMMA_F32_16X16X4_F32 (full fp32), cell update fused in-register.
// K-loop is software-pipelined 2 chunks deep so global loads overlap WMMAs
// (avoids the s_wait_loadcnt 0x0 before every wmma seen in the naive version).
// 512 sequential launches; weights (2MB) and state (512KB) are L2-resident.

typedef __attribute__((ext_vector_type(2))) float v2f;
typedef __attribute__((ext_vector_type(8))) float v8f;

namespace {
constexpr int S_ = 512;
constexpr int B_ = 256;
constexpr int I_ = 256;
constexpr int H_ = 256;

__device__ __forceinline__ float sigmoid_f(float x) {
  return 1.0f / (1.0f + __expf(-x));
}

__device__ __forceinline__ v8f wmma_f32_k4(v2f a, v2f b, v8f c) {
  // 8 args: (neg_a, A, neg_b, B, c_mod, C, reuse_a, reuse_b)
  return __builtin_amdgcn_wmma_f32_16x16x4_f32(
      false, a, false, b, (short)0, c, false, false);
}

// One K=256 panel of the gates GEMM for one wave's 4 gate tiles.
// 2-deep software pipeline: loads for chunk i+2 issue before WMMAs of chunk i.
// MASK: multiply A by m at consumption time (data is 2 iterations old, so the
// wait is already satisfied and the v_mul overlaps outstanding loads).
template <bool MASK>
__device__ __forceinline__ void gemm_panel(
    const float* __restrict__ ap,   // A row pointer (+ko)
    const float* __restrict__ bp0,  // W row pointers for gates i,f,g,o (+ko)
    const float* __restrict__ bp1,
    const float* __restrict__ bp2,
    const float* __restrict__ bp3,
    float m, v8f* __restrict__ acc)
{
  v2f a0 = *(const v2f*)(ap);
  v2f u0 = *(const v2f*)(bp0);
  v2f q0 = *(const v2f*)(bp1);
  v2f w0 = *(const v2f*)(bp2);
  v2f x0 = *(const v2f*)(bp3);
  v2f a1 = *(const v2f*)(ap + 4);
  v2f u1 = *(const v2f*)(bp0 + 4);
  v2f q1 = *(const v2f*)(bp1 + 4);
  v2f w1 = *(const v2f*)(bp2 + 4);
  v2f x1 = *(const v2f*)(bp3 + 4);

#pragma unroll 2
  for (int k = 0; k < 256 - 8; k += 8) {
    // prefetch chunk k+8
    v2f a2 = *(const v2f*)(ap + k + 8);
    v2f u2 = *(const v2f*)(bp0 + k + 8);
    v2f q2 = *(const v2f*)(bp1 + k + 8);
    v2f w2 = *(const v2f*)(bp2 + k + 8);
    v2f x2 = *(const v2f*)(bp3 + k + 8);
    // consume chunk k (loaded 2 iterations ago)
    if (MASK) { a0.x *= m; a0.y *= m; }
    acc[0] = wmma_f32_k4(a0, u0, acc[0]);
    acc[1] = wmma_f32_k4(a0, q0, acc[1]);
    acc[2] = wmma_f32_k4(a0, w0, acc[2]);
    acc[3] = wmma_f32_k4(a0, x0, acc[3]);
    // prefetch chunk k+12
    v2f a3 = *(const v2f*)(ap + k + 12);
    v2f u3 = *(const v2f*)(bp0 + k + 12);
    v2f q3 = *(const v2f*)(bp1 + k + 12);
    v2f w3 = *(const v2f*)(bp2 + k + 12);
    v2f x3 = *(const v2f*)(bp3 + k + 12);
    // consume chunk k+4
    if (MASK) { a1.x *= m; a1.y *= m; }
    acc[0] = wmma_f32_k4(a1, u1, acc[0]);
    acc[1] = wmma_f32_k4(a1, q1, acc[1]);
    acc[2] = wmma_f32_k4(a1, w1, acc[2]);
    acc[3] = wmma_f32_k4(a1, x1, acc[3]);
    // rotate (renamed away by unroll)
    a0 = a2; u0 = u2; q0 = q2; w0 = w2; x0 = x2;
    a1 = a3; u1 = u3; q1 = q3; w1 = w3; x1 = x3;
  }
  // epilogue: chunks 248, 252
  if (MASK) { a0.x *= m; a0.y *= m; }
  acc[0] = wmma_f32_k4(a0, u0, acc[0]);
  acc[1] = wmma_f32_k4(a0, q0, acc[1]);
  acc[2] = wmma_f32_k4(a0, w0, acc[2]);
  acc[3] = wmma_f32_k4(a0, x0, acc[3]);
  if (MASK) { a1.x *= m; a1.y *= m; }
  acc[0] = wmma_f32_k4(a1, u1, acc[0]);
  acc[1] = wmma_f32_k4(a1, q1, acc[1]);
  acc[2] = wmma_f32_k4(a1, w1, acc[2]);
  acc[3] = wmma_f32_k4(a1, x1, acc[3]);
}
} // namespace

// Grid: 64 blocks x 128 threads (4 waves). blockIdx&15 -> hidden tile,
// (blockIdx>>4)*4 + wave -> batch tile. Each wave accumulates the four gate
// tiles (i,f,g,o) for its (batch, hidden) tile and fuses the cell update.
__global__ __launch_bounds__(128) void lstm_step_kernel(
    const float* __restrict__ x_t,     // (B, I)  inputs at step t
    const float* __restrict__ h_prev,  // (B, H)  previous hidden (or h0)
    const float* __restrict__ c_prev,  // (B, H)  previous cell   (or c0)
    const float* __restrict__ done_t,  // (B,)    done flags at step t
    const float* __restrict__ W_ih,    // (4H, I) row-major
    const float* __restrict__ W_hh,    // (4H, H) row-major
    const float* __restrict__ b_ih,    // (4H,)
    const float* __restrict__ b_hh,    // (4H,)
    float* __restrict__ h_out,         // (B, H)  hs[t] (in d_out)
    float* __restrict__ c_out,         // (B, H)  cell state (in d_ws)
    float* __restrict__ hT_out,        // (B, H)  final-h tail of d_out
    float* __restrict__ cT_out,        // (B, H)  final-c tail of d_out
    int is_last)
{
  const int lane = threadIdx.x & 31;
  const int wave = threadIdx.x >> 5;
  const int th   = blockIdx.x & 15;                 // hidden tile
  const int tb   = ((blockIdx.x >> 4) << 2) + wave; // batch tile
  const int b0   = tb * 16;
  const int n0   = th * 16;
  const int lr   = lane & 15;
  const int ko   = (lane >> 4) << 1;  // K sub-offset: 0 (lanes 0-15) / 2 (16-31)

  // Accumulators i,f,g,o — C/D layout: col N = lane&15 for all 8 row-VGPRs,
  // so the per-column bias broadcasts across the whole v8f.
  v8f acc[4];
  const int cbase = n0 + lr;
#pragma unroll
  for (int g = 0; g < 4; ++g) {
    const int cn = g * H_ + cbase;  // column in 4H space
    const float bias = b_ih[cn] + b_hh[cn];
    acc[g] = (v8f){bias, bias, bias, bias, bias, bias, bias, bias};
  }

  // A-matrix fp32 16x4 layout: lane L<16 holds row b0+L, K = k..k+1;
  // lane L>=16 holds row b0+(L-16), K = k+2..k+3  -> one float2 per chunk.
  const int ar = b0 + lr;
  const float m = 1.0f - done_t[ar];  // reset mask for this A row

  // ---- K panel 0: x_t @ W_ih^T ----
  gemm_panel<false>(
      x_t + (size_t)ar * I_ + ko,
      W_ih + (size_t)(0 * H_ + cbase) * I_ + ko,
      W_ih + (size_t)(1 * H_ + cbase) * I_ + ko,
      W_ih + (size_t)(2 * H_ + cbase) * I_ + ko,
      W_ih + (size_t)(3 * H_ + cbase) * I_ + ko,
      0.0f, acc);

  // ---- K panel 1: (mask * h_prev) @ W_hh^T ----
  gemm_panel<true>(
      h_prev + (size_t)ar * H_ + ko,
      W_hh + (size_t)(0 * H_ + cbase) * H_ + ko,
      W_hh + (size_t)(1 * H_ + cbase) * H_ + ko,
      W_hh + (size_t)(2 * H_ + cbase) * H_ + ko,
      W_hh + (size_t)(3 * H_ + cbase) * H_ + ko,
      m, acc);

  // ---- fused cell update (C/D layout: VGPR e -> row b0 + e + 8*(lane>=16)) ----
  const int rbase = b0 + ((lane >> 4) << 3);
  const int hc = cbase;
#pragma unroll
  for (int e = 0; e < 8; ++e) {
    const int rr = rbase + e;
    const float me = 1.0f - done_t[rr];
    const float cp = c_prev[(size_t)rr * H_ + hc] * me;
    const float ig = sigmoid_f(acc[0][e]);
    const float fg = sigmoid_f(acc[1][e]);
    const float gg = tanhf(acc[2][e]);
    const float og = sigmoid_f(acc[3][e]);
    const float cnew = fg * cp + ig * gg;
    const float hnew = og * tanhf(cnew);
    h_out[(size_t)rr * H_ + hc] = hnew;
    c_out[(size_t)rr * H_ + hc] = cnew;
    if (is_last) {
      hT_out[(size_t)rr * H_ + hc] = hnew;
      cT_out[(size_t)rr * H_ + hc] = cnew;
    }
  }
}

extern "C" void kernel_launch(void* const* d_in, const int* in_sizes, int n_in,
                              void* d_out, int out_size, void* d_ws, size_t ws_size,
                              hipStream_t stream) {
  const float* inputs = (const float*)d_in[0]; // (S, B, I)
  const float* h0     = (const float*)d_in[1]; // (1, B, H)
  const float* c0     = (const float*)d_in[2]; // (1, B, H)
  const float* done   = (const float*)d_in[3]; // (S, B)
  const float* W_ih   = (const float*)d_in[4]; // (4H, I)
  const float* W_hh   = (const float*)d_in[5]; // (4H, H)
  const float* b_ih   = (const float*)d_in[6]; // (4H,)
  const float* b_hh   = (const float*)d_in[7]; // (4H,)

  float* out  = (float*)d_out;
  float* hs   = out;                               // (S*B, H)
  float* hT   = out + (size_t)S_ * B_ * H_;        // (B, H)
  float* cT   = hT + (size_t)B_ * H_;              // (B, H)
  float* c_ws = (float*)d_ws;                      // (B, H) cell state

  for (int t = 0; t < S_; ++t) {
    const float* x_t = inputs + (size_t)t * B_ * I_;
    const float* hp  = (t == 0) ? h0 : hs + (size_t)(t - 1) * B_ * H_;
    const float* cp  = (t == 0) ? c0 : c_ws;
    const float* dt  = done + (size_t)t * B_;
    float* ho = hs + (size_t)t * B_ * H_;
    lstm_step_kernel<<<dim3(64), dim3(128), 0, stream>>>(
        x_t, hp, cp, dt, W_ih, W_hh, b_ih, b_hh,
        ho, c_ws, hT, cT, (t == S_ - 1) ? 1 : 0);
  }
}